// StandardPershomReadout_31705448579349
// MI455X (gfx1250) — compile-verified
//
#include <hip/hip_runtime.h>
#include <hip/hip_bf16.h>

typedef __attribute__((ext_vector_type(2))) float v2f;
typedef __attribute__((ext_vector_type(8))) float v8f;

__device__ __forceinline__ float fast_rcp(float x) {
    // v_rcp_f32 (~1 ulp) + one Newton-Raphson step -> ~0.5 ulp
    float r = __builtin_amdgcn_rcpf(x);
    return r * (2.0f - x * r);
}

// f(d) = 1/(1+d) - 1/(1+| |r| - d |)
__device__ __forceinline__ float rhat(float d, float rabs) {
    return fast_rcp(1.0f + d) - fast_rcp(1.0f + fabsf(rabs - d));
}

// Per-wave accumulation: lane owns two centers; points broadcast across wave.
template <int D>
__device__ __forceinline__ void accum_seg(const float* __restrict__ pts,
                                          const float* __restrict__ mask,
                                          float cx0, float cy0, float cx1, float cy1,
                                          float rabs, int P, int lane, int wv,
                                          float& acc0, float& acc1) {
    for (int ch = wv; ch * 32 < P; ch += 8) {
        int p = ch * 32 + lane;
        float px = 0.0f, py = 0.0f, m = 0.0f;
        if (p < P) {
            if (D == 2) {
                float2 xy = reinterpret_cast<const float2*>(pts)[p];
                px = xy.x; py = xy.y;
            } else {
                px = pts[p];
            }
            m = mask[p];
        }
#pragma unroll 8
        for (int j = 0; j < 32; ++j) {
            float xj = __shfl(px, j, 32);
            float yj = (D == 2) ? __shfl(py, j, 32) : 0.0f;
            float mj = __shfl(m, j, 32);

            float d0 = fabsf(xj - cx0);
            float d1 = fabsf(xj - cx1);
            if (D == 2) {
                d0 += fabsf(yj - cy0);
                d1 += fabsf(yj - cy1);
            }
            acc0 = fmaf(mj, rhat(d0, rabs), acc0);
            acc1 = fmaf(mj, rhat(d1, rabs), acc1);
        }
    }
}

__global__ __launch_bounds__(256) void pershom_readout_kernel(
    const float* __restrict__ h0,  const float* __restrict__ m0,
    const float* __restrict__ h0e, const float* __restrict__ m0e,
    const float* __restrict__ h1e, const float* __restrict__ m1e,
    const float* __restrict__ c0,  const float* __restrict__ r0,
    const float* __restrict__ c0e, const float* __restrict__ r0e,
    const float* __restrict__ c1e, const float* __restrict__ r1e,
    float* __restrict__ out, int P0, int PE, int K) {
    __shared__ float red[8][64];

    const int b = blockIdx.x;
    const int seg = blockIdx.y;
    const int lane = threadIdx.x & 31;
    const int wv = threadIdx.x >> 5;

    const float* pts;
    const float* mask;
    const float* ctr;
    float rr;
    int P;
    if (seg == 0) {
        pts = h0 + (size_t)b * P0 * 2; mask = m0 + (size_t)b * P0;
        ctr = c0; rr = r0[0]; P = P0;
    } else if (seg == 1) {
        pts = h0e + (size_t)b * PE;    mask = m0e + (size_t)b * PE;
        ctr = c0e; rr = r0e[0]; P = PE;
    } else {
        pts = h1e + (size_t)b * PE;    mask = m1e + (size_t)b * PE;
        ctr = c1e; rr = r1e[0]; P = PE;
    }
    const float rabs = fabsf(rr);

    float acc0 = 0.0f, acc1 = 0.0f;
    if (seg == 0) {
        float cx0 = ctr[2 * lane],        cy0 = ctr[2 * lane + 1];
        float cx1 = ctr[2 * (lane + 32)], cy1 = ctr[2 * (lane + 32) + 1];
        accum_seg<2>(pts, mask, cx0, cy0, cx1, cy1, rabs, P, lane, wv, acc0, acc1);
    } else {
        float cx0 = ctr[lane], cx1 = ctr[lane + 32];
        accum_seg<1>(pts, mask, cx0, 0.0f, cx1, 0.0f, rabs, P, lane, wv, acc0, acc1);
    }

    // Per-wave partials: red[wave][center]
    red[wv][lane]      = acc0;
    red[wv][lane + 32] = acc1;
    __syncthreads();

    // Reduce 8 wave-partials per center with v_wmma_f32_16x16x4_f32:
    // A = ones(16x4), B rows = 4 wave-partial rows, two chained WMMAs sum all 8.
    // With A==1 the result is the column sum regardless of which (VGPR,half)
    // slot holds which B row; column index = lane%16.
    if (wv == 0) {
        v2f a = {1.0f, 1.0f};  // ones A-matrix (all lanes, both K slots)
#pragma unroll
        for (int t = 0; t < 4; ++t) {
            const int col = t * 16 + (lane & 15);
            const int rb = (lane < 16) ? 0 : 2;
            v8f c = {};
            v2f bm;
            bm[0] = red[rb + 0][col];
            bm[1] = red[rb + 1][col];
            c = __builtin_amdgcn_wmma_f32_16x16x4_f32(false, a, false, bm,
                                                      (short)0, c, false, false);
            bm[0] = red[rb + 4][col];
            bm[1] = red[rb + 5][col];
            c = __builtin_amdgcn_wmma_f32_16x16x4_f32(false, a, false, bm,
                                                      (short)0, c, false, false);
            if (lane < 16) {
                out[(size_t)b * (3 * K) + seg * K + t * 16 + lane] = c[0];
            }
        }
    }
}

extern "C" void kernel_launch(void* const* d_in, const int* in_sizes, int n_in,
                              void* d_out, int out_size, void* d_ws, size_t ws_size,
                              hipStream_t stream) {
    const float* h0  = (const float*)d_in[0];
    const float* m0  = (const float*)d_in[1];
    const float* h0e = (const float*)d_in[2];
    const float* m0e = (const float*)d_in[3];
    const float* h1e = (const float*)d_in[4];
    const float* m1e = (const float*)d_in[5];
    const float* c0  = (const float*)d_in[6];
    const float* r0  = (const float*)d_in[7];
    const float* c0e = (const float*)d_in[8];
    const float* r0e = (const float*)d_in[9];
    const float* c1e = (const float*)d_in[10];
    const float* r1e = (const float*)d_in[11];

    const int K  = in_sizes[6] / 2;        // centers_0 is (K,2) -> 64
    const int B  = out_size / (3 * K);     // output (B, 3K)
    const int P0 = in_sizes[1] / B;        // mask_0 is (B,P0)
    const int PE = in_sizes[3] / B;        // mask_0_ess is (B,PE)

    dim3 grid(B, 3), block(256);
    pershom_readout_kernel<<<grid, block, 0, stream>>>(
        h0, m0, h0e, m0e, h1e, m1e, c0, r0, c0e, r0e, c1e, r1e,
        (float*)d_out, P0, PE, K);
}